// LocalizationLoss_32195074851366
// MI455X (gfx1250) — compile-verified
//
#include <hip/hip_runtime.h>
#include <math.h>

typedef __attribute__((ext_vector_type(2))) float v2f;
typedef __attribute__((ext_vector_type(8))) float v8f;

#define NB    8
#define HH    256
#define WW    256
#define PX    (HH * WW)        // 65536 pixels per batch
#define NTILES (PX / 16)       // 4096 pixel tiles per batch
#define NN    100
#define NCT   7                // column tiles (ceil(100/16))
#define NPAD  (NCT * 16)       // 112
#define BIGV  1e30f
#define CONF_TH 0.5f

// workspace layout in floats:
//   [0, NB*NPAD)        : per (b,n) running min of (a^2 - 2 a.b)
//   [.., +NB*NN)        : confW (exp(-u) of first 100 confident pixels, row-major)
//   [.., +NB*NN)        : nonW  (exp(-u) of first 100 non-confident pixels)
//   then NB uints       : nc per batch
#define WS_MINQ  0
#define WS_CONFW (NB * NPAD)
#define WS_NONW  (WS_CONFW + NB * NN)
#define WS_NC    (WS_NONW + NB * NN)

__device__ __forceinline__ float atomicMinFloat(float* addr, float val) {
  // sign-split trick: valid because slots are initialized to +BIGV
  if (val >= 0.f)
    return __int_as_float(atomicMin((int*)addr, __float_as_int(val)));
  else
    return __uint_as_float(atomicMax((unsigned int*)addr, __float_as_uint(val)));
}

__global__ __launch_bounds__(256) void ll_init_kernel(float* ws) {
  int i = blockIdx.x * blockDim.x + threadIdx.x;
  if (i < NB * NPAD) ws[WS_MINQ + i] = BIGV;
  if (i < NB * NN) {
    ws[WS_CONFW + i] = 1.f;
    ws[WS_NONW + i] = 1.f;
  }
  if (i < NB) ((unsigned int*)(ws + WS_NC))[i] = 0u;
}

// ---------------------------------------------------------------------------
// WMMA pairwise-distance + min kernel.
// A row (pixel M), confident:     [ax, ay, ax^2+ay^2, 0]
// A row (pixel M), non-confident: [ 0,  0, BIGV,      0]   <- mask folded in
// B col (target N):               [-2bx, -2by, 1, 0]
// => D[M][N] = a^2 - 2 a.b  (confident)   or exactly BIGV (non-confident).
// b^2 is added in finalize; sqrt/min are monotone so ordering is preserved.
// A layout (f32 16x4): lanes 0-15 -> {K0,K1} of row M=lane&15; lanes 16-31 -> {K2,K3}.
// D layout: VGPR r, lanes 0-15: M=r, N=lane; lanes 16-31: M=r+8, N=lane-16.
// Per column tile we keep ONE accumulator and min-tree the 8 D elements so the
// backend can form v_min3_num_f32 (8 inputs -> ~4 ops instead of 8).
// ---------------------------------------------------------------------------
__global__ __launch_bounds__(256) void ll_distmin_kernel(
    const float* __restrict__ loc,   // [B,H,W,2]
    const float* __restrict__ prob,  // [B,1,H,W]
    const float* __restrict__ tl,    // [B,N,2]
    float* __restrict__ ws) {
  float* minq = ws + WS_MINQ;
  unsigned int* ncArr = (unsigned int*)(ws + WS_NC);

  const int b = blockIdx.y;
  const int lane = threadIdx.x & 31;
  const int wave = threadIdx.x >> 5;
  const int wavesPerBlock = blockDim.x >> 5;
  const int waveGlobal = blockIdx.x * wavesPerBlock + wave;
  const int nWaves = gridDim.x * wavesPerBlock;
  const int col = lane & 15;
  const bool hiHalf = lane >= 16;

  // preload B fragments for the 7 column tiles (N padded with zeros)
  v2f Bf[NCT];
#pragma unroll
  for (int j = 0; j < NCT; ++j) {
    int n = j * 16 + col;
    float bx = 0.f, by = 0.f;
    if (n < NN) {
      bx = tl[((size_t)b * NN + n) * 2 + 0];
      by = tl[((size_t)b * NN + n) * 2 + 1];
    }
    if (!hiHalf) { Bf[j].x = -2.f * bx; Bf[j].y = -2.f * by; }
    else         { Bf[j].x = 1.0f;      Bf[j].y = 0.0f;      }
  }

  float acc[NCT];
#pragma unroll
  for (int j = 0; j < NCT; ++j) acc[j] = BIGV;

  unsigned int confCount = 0;  // wave-uniform
  const float2* locb = (const float2*)(loc + (size_t)b * PX * 2);
  const float* probb = prob + (size_t)b * PX;

  for (int tile = waveGlobal; tile < NTILES; tile += nWaves) {
    const int m = tile * 16 + col;  // pixel index (both halves load row m)
    const float2 xy = locb[m];      // coalesced b64 load
    const float pv = probb[m];
    const bool conf = pv > CONF_TH;

    // wave-uniform confident count for this tile (low 16 ballot bits = tile)
    confCount += __popc(((unsigned int)__ballot(conf)) & 0xFFFFu);

    // Build A fragment with the mask folded in (2-3 cndmasks total).
    v2f Af;
    if (!hiHalf) {
      Af.x = conf ? xy.x : 0.f;
      Af.y = conf ? xy.y : 0.f;
    } else {
      Af.x = conf ? (xy.x * xy.x + xy.y * xy.y) : BIGV;
      Af.y = 0.f;
    }

#pragma unroll
    for (int j = 0; j < NCT; ++j) {
      v8f c = {};
      v8f d = __builtin_amdgcn_wmma_f32_16x16x4_f32(
          false, Af, false, Bf[j], (short)0, c, false, false);
      // 8-input min tree (encourages v_min3_num_f32 formation)
      float m01 = fminf(d[0], d[1]);
      float m012 = fminf(m01, d[2]);
      float m34 = fminf(d[3], d[4]);
      float m345 = fminf(m34, d[5]);
      float m67 = fminf(d[6], d[7]);
      float mt = fminf(fminf(m012, m345), m67);
      acc[j] = fminf(acc[j], mt);
    }
  }

  // combine the two lane halves (lane l and l+16 share column N=l), atomic-min.
#pragma unroll
  for (int j = 0; j < NCT; ++j) {
    float m = fminf(acc[j], __shfl_xor(acc[j], 16, 32));
    const int n = j * 16 + col;
    if (!hiHalf && n < NN) atomicMinFloat(&minq[b * NPAD + n], m);
  }

  if (lane == 0 && confCount) atomicAdd(&ncArr[b], confCount);
}

// ---------------------------------------------------------------------------
// One wave per batch: row-major scan with ballot prefix to capture exp(-u) of
// the first 100 confident and first 100 non-confident pixels.
// ---------------------------------------------------------------------------
__global__ __launch_bounds__(32) void ll_weights_kernel(
    const float* __restrict__ unc,   // [B,H,W,1]
    const float* __restrict__ prob,  // [B,1,H,W]
    float* __restrict__ ws) {
  const int b = blockIdx.x;
  const int lane = threadIdx.x;
  float* confW = ws + WS_CONFW + b * NN;
  float* nonW  = ws + WS_NONW + b * NN;
  const float* probb = prob + (size_t)b * PX;
  const float* uncb  = unc + (size_t)b * PX;

  unsigned int cDone = 0, nDone = 0;  // wave-uniform running ordinals
  const unsigned int below = (lane == 0) ? 0u : ((1u << lane) - 1u);

  for (int base = 0; base < PX; base += 32) {
    const int p = base + lane;
    const float pv = probb[p];
    const bool conf = pv > CONF_TH;
    const unsigned int cm = (unsigned int)__ballot(conf);
    const unsigned int myC = cDone + __popc(cm & below);
    const unsigned int myN = nDone + __popc((~cm) & below);
    if (conf && myC < NN) confW[myC] = __expf(-uncb[p]);
    if (!conf && myN < NN) nonW[myN] = __expf(-uncb[p]);
    cDone += __popc(cm);
    nDone += 32u - __popc(cm);
    if (cDone >= NN && nDone >= NN) break;  // uniform exit
  }
}

// ---------------------------------------------------------------------------
// Single-wave finalize across all batches.
// ---------------------------------------------------------------------------
__global__ __launch_bounds__(32) void ll_finalize_kernel(
    const float* __restrict__ tl,  // [B,N,2]
    const float* __restrict__ ws,
    float* __restrict__ out) {
  const int lane = threadIdx.x;
  const float* minq = ws + WS_MINQ;
  const unsigned int* ncArr = (const unsigned int*)(ws + WS_NC);
  const unsigned int incl = (2u << lane) - 1u;  // lane 31 wraps to 0xFFFFFFFF

  float total = 0.f;
  for (int b = 0; b < NB; ++b) {
    const unsigned int nc = ncArr[b];
    const float* confW = ws + WS_CONFW + b * NN;
    const float* nonW  = ws + WS_NONW + b * NN;
    float sum = 0.f;
    int nv = 0;
    for (int base = 0; base < NN; base += 32) {
      const int n = base + lane;
      float bx = -1.f, by = -1.f;
      if (n < NN) {
        bx = tl[((size_t)b * NN + n) * 2 + 0];
        by = tl[((size_t)b * NN + n) * 2 + 1];
      }
      const bool valid = (n < NN) && (bx >= 0.f) && (by >= 0.f);
      const unsigned int vm = (unsigned int)__ballot(valid);
      if (valid) {
        const unsigned int rank = (unsigned int)(nv + __popc(vm & incl) - 1);
        const float w = (rank < nc) ? confW[rank] : nonW[rank - nc];
        const float b2 = bx * bx + by * by;
        const float q = minq[b * NPAD + n];
        const float d = sqrtf(fmaxf(q + b2, 0.f));
        sum += d * w;
      }
      nv += __popc(vm);
    }
    for (int off = 16; off > 0; off >>= 1) sum += __shfl_xor(sum, off, 32);
    const float loss = sum / (float)((nv > 0) ? nv : 1);
    const float res = (nv == 0) ? 0.f : ((nc == 0u) ? 10.0f : loss);
    total += res;
  }
  if (lane == 0) out[0] = total / (float)NB;
}

extern "C" void kernel_launch(void* const* d_in, const int* in_sizes, int n_in,
                              void* d_out, int out_size, void* d_ws, size_t ws_size,
                              hipStream_t stream) {
  const float* loc  = (const float*)d_in[0];  // [B,H,W,2]
  const float* unc  = (const float*)d_in[1];  // [B,H,W,1]
  const float* tl   = (const float*)d_in[2];  // [B,N,2]
  const float* prob = (const float*)d_in[3];  // [B,1,H,W]
  float* ws = (float*)d_ws;
  float* out = (float*)d_out;

  ll_init_kernel<<<4, 256, 0, stream>>>(ws);

  dim3 grid(32, NB);  // 32 blocks x 8 waves = 256 waves per batch, 16 tiles/wave
  ll_distmin_kernel<<<grid, 256, 0, stream>>>(loc, prob, tl, ws);

  ll_weights_kernel<<<NB, 32, 0, stream>>>(unc, prob, ws);

  ll_finalize_kernel<<<1, 32, 0, stream>>>(tl, ws, out);
}